// BitSwapWrapper_89627377533020
// MI455X (gfx1250) — compile-verified
//
#include <hip/hip_runtime.h>

typedef float v2f __attribute__((ext_vector_type(2)));
typedef float v4f __attribute__((ext_vector_type(4)));
typedef float v8f __attribute__((ext_vector_type(8)));

#define B_DIM 4096
#define F_DIM 16384
#define O_DIM 256

#define BM 64
#define BN 64
#define BK 32
#define LDA (BK + 4)   // 36 floats: 16B-aligned rows, conflict-free b64 frag reads
#define LDB (BN + 4)   // 68 floats
#define NTHREADS 128
#define NTILES (F_DIM / BK)   // 512

#if defined(__gfx1250__) && __has_builtin(__builtin_amdgcn_global_load_async_to_lds_b128) && __has_builtin(__builtin_amdgcn_s_wait_asynccnt)
#define USE_ASYNC 1
#define NBUF 3
typedef int int4v __attribute__((ext_vector_type(4)));
typedef __attribute__((address_space(1))) int4v* gas_ptr;       // global <4 x i32>*
typedef __attribute__((address_space(3))) void*  lds_voidp;     // LDS void*
typedef __attribute__((address_space(3))) int4v* lds_ptr;       // LDS <4 x i32>*
#else
#define USE_ASYNC 0
#define NBUF 2
#endif

// ---------------------------------------------------------------------------
// Kernel 1: per-row perturbation scalar u[i] = (bitflip(g) - g) * coeff
// ---------------------------------------------------------------------------
__global__ void __launch_bounds__(256)
u_kernel(const float* __restrict__ x, const int* __restrict__ idx,
         const int* __restrict__ bitpos, const float* __restrict__ coeff,
         float* __restrict__ u) {
    int i = blockIdx.x * blockDim.x + threadIdx.x;
    if (i < B_DIM) {
        int j = idx[i];
        float g = x[(size_t)i * F_DIM + j];
        int gi = __float_as_int(g) ^ (1 << bitpos[i]);
        u[i] = (__int_as_float(gi) - g) * coeff[0];
    }
}

// ---------------------------------------------------------------------------
// Kernel 2: out = x @ W + b + u[i] * W[idx[i], :]   (fp32 WMMA 16x16x4)
// 128 threads = 4 waves; block tile 64x64, K tile 32; wave tile 32x32.
// Triple-buffered LDS, 2-tile-deep async global->LDS pipeline.
// ---------------------------------------------------------------------------
__global__ void __launch_bounds__(NTHREADS)
gemm_kernel(const float* __restrict__ x, const float* __restrict__ W,
            const float* __restrict__ bias, const float* __restrict__ u,
            const int* __restrict__ idx, float* __restrict__ out) {

    __shared__ float As[NBUF][BM * LDA];
    __shared__ float Bs[NBUF][BK * LDB];

    const int tid      = threadIdx.x;
    const int lane     = tid & 31;
    const int wave     = tid >> 5;
    const int wm       = (wave & 1) * 32;
    const int wn       = (wave >> 1) * 32;
    const int laneHalf = lane >> 4;
    const int laneLow  = lane & 15;

    const int rowBase = blockIdx.y * BM;
    const int colBase = blockIdx.x * BN;

    v8f acc[2][2] = {};

    // Global->LDS mapping: A tile 64x32 (8 float4/row), B tile 32x64 (16 float4/row)
    const int aRow0 = tid >> 3;         // 0..15 (+16 strides)
    const int aCol  = (tid & 7) * 4;
    const int bRow0 = tid >> 4;         // 0..7 (+8 strides)
    const int bCol  = (tid & 15) * 4;

#if USE_ASYNC
    auto issue_tile = [&](int k0, int buf) {
        #pragma unroll
        for (int r = 0; r < 4; ++r) {
            int row = aRow0 + r * 16;
            __builtin_amdgcn_global_load_async_to_lds_b128(
                (gas_ptr)(unsigned long long)&x[(size_t)(rowBase + row) * F_DIM + k0 + aCol],
                (lds_ptr)(lds_voidp)&As[buf][row * LDA + aCol], 0, 0);
        }
        #pragma unroll
        for (int r = 0; r < 4; ++r) {
            int row = bRow0 + r * 8;
            __builtin_amdgcn_global_load_async_to_lds_b128(
                (gas_ptr)(unsigned long long)&W[(size_t)(k0 + row) * O_DIM + colBase + bCol],
                (lds_ptr)(lds_voidp)&Bs[buf][row * LDB + bCol], 0, 0);
        }
    };
#else
    v4f ra[4], rb[4];
    auto gload = [&](int k0) {
        #pragma unroll
        for (int r = 0; r < 4; ++r)
            ra[r] = *(const v4f*)&x[(size_t)(rowBase + aRow0 + r * 16) * F_DIM + k0 + aCol];
        #pragma unroll
        for (int r = 0; r < 4; ++r)
            rb[r] = *(const v4f*)&W[(size_t)(k0 + bRow0 + r * 8) * O_DIM + colBase + bCol];
    };
    auto lstore = [&](int buf) {
        #pragma unroll
        for (int r = 0; r < 4; ++r)
            *(v4f*)&As[buf][(aRow0 + r * 16) * LDA + aCol] = ra[r];
        #pragma unroll
        for (int r = 0; r < 4; ++r)
            *(v4f*)&Bs[buf][(bRow0 + r * 8) * LDB + bCol] = rb[r];
    };
#endif

    auto compute = [&](int buf) {
        #pragma unroll
        for (int kk = 0; kk < BK; kk += 4) {
            const int ka = kk + 2 * laneHalf;
            v2f a0 = *(const v2f*)&As[buf][(wm + laneLow) * LDA + ka];
            v2f a1 = *(const v2f*)&As[buf][(wm + 16 + laneLow) * LDA + ka];
            v2f b0, b1;
            b0.x = Bs[buf][ka * LDB + wn + laneLow];
            b0.y = Bs[buf][(ka + 1) * LDB + wn + laneLow];
            b1.x = Bs[buf][ka * LDB + wn + 16 + laneLow];
            b1.y = Bs[buf][(ka + 1) * LDB + wn + 16 + laneLow];
            acc[0][0] = __builtin_amdgcn_wmma_f32_16x16x4_f32(
                false, a0, false, b0, (short)0, acc[0][0], false, false);
            acc[0][1] = __builtin_amdgcn_wmma_f32_16x16x4_f32(
                false, a0, false, b1, (short)0, acc[0][1], false, false);
            acc[1][0] = __builtin_amdgcn_wmma_f32_16x16x4_f32(
                false, a1, false, b0, (short)0, acc[1][0], false, false);
            acc[1][1] = __builtin_amdgcn_wmma_f32_16x16x4_f32(
                false, a1, false, b1, (short)0, acc[1][1], false, false);
        }
    };

#if USE_ASYNC
    // 2-tile-deep software pipeline over 3 LDS buffers.
    issue_tile(0, 0);
    issue_tile(BK, 1);
    int bufc = 0;   // buffer holding tile t
    int bufi = 2;   // buffer to receive tile t+2
    for (int t = 0; t < NTILES; ++t) {
        if (t + 1 < NTILES) {
            __builtin_amdgcn_s_wait_asynccnt(8);   // tile t landed; t+1 may be in flight
        } else {
            __builtin_amdgcn_s_wait_asynccnt(0);
        }
        __syncthreads();   // all waves: tile t visible, compute(t-1) finished everywhere
        if (t + 2 < NTILES) issue_tile((t + 2) * BK, bufi);
        compute(bufc);
        bufc = (bufc == NBUF - 1) ? 0 : bufc + 1;
        bufi = (bufi == NBUF - 1) ? 0 : bufi + 1;
    }
#else
    gload(0);
    lstore(0);
    __syncthreads();
    for (int t = 0; t < NTILES; ++t) {
        const int cur = t & 1;
        if (t + 1 < NTILES) gload((t + 1) * BK);
        compute(cur);
        if (t + 1 < NTILES) lstore(cur ^ 1);
        __syncthreads();
    }
#endif

    // Epilogue: + bias + u[row] * W[idx[row], col]
    // C/D layout: VGPR v of lane l = (row v + 8*(l/16), col l%16)
    #pragma unroll
    for (int mi = 0; mi < 2; ++mi) {
        #pragma unroll
        for (int v = 0; v < 8; ++v) {
            int grow = rowBase + wm + mi * 16 + v + 8 * laneHalf;
            float uu = u[grow];
            int   id = idx[grow];
            #pragma unroll
            for (int ni = 0; ni < 2; ++ni) {
                int gcol = colBase + wn + ni * 16 + laneLow;
                float val = acc[mi][ni][v];
                val += bias[gcol] + uu * W[(size_t)id * O_DIM + gcol];
                out[(size_t)grow * O_DIM + gcol] = val;
            }
        }
    }
}

// ---------------------------------------------------------------------------
extern "C" void kernel_launch(void* const* d_in, const int* in_sizes, int n_in,
                              void* d_out, int out_size, void* d_ws, size_t ws_size,
                              hipStream_t stream) {
    const float* x      = (const float*)d_in[0];
    const float* W      = (const float*)d_in[1];
    const float* bias   = (const float*)d_in[2];
    const float* coeff  = (const float*)d_in[3];
    const int*   idx    = (const int*)d_in[4];
    const int*   bitpos = (const int*)d_in[5];
    float*       out    = (float*)d_out;
    float*       u      = (float*)d_ws;   // 4096 floats of scratch

    u_kernel<<<(B_DIM + 255) / 256, 256, 0, stream>>>(x, idx, bitpos, coeff, u);

    dim3 grid(O_DIM / BN, B_DIM / BM);    // 4 x 64 = 256 workgroups
    gemm_kernel<<<grid, NTHREADS, 0, stream>>>(x, W, bias, u, idx, out);
}